// SCSA_33724083208493
// MI455X (gfx1250) — compile-verified
//
#include <hip/hip_runtime.h>
#include <math.h>

typedef float v2f __attribute__((ext_vector_type(2)));
typedef float v8f __attribute__((ext_vector_type(8)));

// ---------------------------------------------------------------------------
// Generic f32 WMMA GEMM:  out[b] = A[b] (MxK) @ X[b] (KxNcols)  (+bias, *scale)
// One wave per 16x64 output tile, K-step 8 (8 WMMAs/iter). Templated epilogue:
//   ACCUM   : split-K partial -> atomicAdd (bias must be off)
//   HASBIAS : per-row bias added on store
//   ACONTIG : A column-stride == 1 -> 64-bit A-fragment loads
// ---------------------------------------------------------------------------
template<bool ACCUM, bool HASBIAS, bool ACONTIG>
__global__ __launch_bounds__(32)
void wmma_gemm_f32(const float* __restrict__ A, long long aBS, int aRS, int aCS,
                   const float* __restrict__ X, long long xBS,
                   const float* __restrict__ bias,
                   float* __restrict__ out, long long oBS,
                   int Kdim, int Ncols, int nGroups, int splitK, float outScale)
{
    const int b  = blockIdx.x;
    const int m0 = blockIdx.y * 16;
    const int zg = blockIdx.z;
    const int ng = zg % nGroups;
    const int ks = zg / nGroups;
    const int n0 = ng * 64;
    const int kChunk = Kdim / splitK;
    const int kStart = ks * kChunk;
    const int kEnd   = kStart + kChunk;

    const int lane  = threadIdx.x & 31;
    const int laneL = lane & 15;
    const int hi    = lane >> 4;

    const float* Ab = A + (long long)b * aBS;
    const float* Xb = X + (long long)b * xBS;

    v8f acc0 = {}, acc1 = {}, acc2 = {}, acc3 = {};
    const long long aRow = (long long)(m0 + laneL) * aRS;

#define GEMM_STEP(KK)                                                          \
    {                                                                          \
        const int ka = (KK) + 2 * hi;                                          \
        v2f a;                                                                 \
        if (ACONTIG) {                                                         \
            a = *(const v2f*)(Ab + aRow + ka);                                 \
        } else {                                                               \
            a.x = Ab[aRow + (long long)ka * aCS];                              \
            a.y = Ab[aRow + (long long)(ka + 1) * aCS];                        \
        }                                                                      \
        const float* x0 = Xb + (long long)ka * Ncols + n0 + laneL;             \
        const float* x1 = x0 + Ncols;                                          \
        v2f b0; b0.x = x0[0];  b0.y = x1[0];                                   \
        v2f b1; b1.x = x0[16]; b1.y = x1[16];                                  \
        v2f b2; b2.x = x0[32]; b2.y = x1[32];                                  \
        v2f b3; b3.x = x0[48]; b3.y = x1[48];                                  \
        acc0 = __builtin_amdgcn_wmma_f32_16x16x4_f32(false, a, false, b0, (short)0, acc0, false, false); \
        acc1 = __builtin_amdgcn_wmma_f32_16x16x4_f32(false, a, false, b1, (short)0, acc1, false, false); \
        acc2 = __builtin_amdgcn_wmma_f32_16x16x4_f32(false, a, false, b2, (short)0, acc2, false, false); \
        acc3 = __builtin_amdgcn_wmma_f32_16x16x4_f32(false, a, false, b3, (short)0, acc3, false, false); \
    }

    for (int kk = kStart; kk < kEnd; kk += 8) {
        GEMM_STEP(kk);
        GEMM_STEP(kk + 4);
    }
#undef GEMM_STEP

    v8f accs[4] = {acc0, acc1, acc2, acc3};
    float* Ob = out + (long long)b * oBS;

    float brow[8];
    if (HASBIAS) {
#pragma unroll
        for (int r = 0; r < 8; ++r) brow[r] = bias[m0 + r + 8 * hi];
    }

#pragma unroll
    for (int t = 0; t < 4; ++t) {
#pragma unroll
        for (int r = 0; r < 8; ++r) {
            const int orow = m0 + r + 8 * hi;
            const int ocol = n0 + t * 16 + laneL;
            const long long off = (long long)orow * Ncols + ocol;
            float val = accs[t][r] * outScale;
            if (ACCUM) {
                atomicAdd(&Ob[off], val);
            } else {
                if (HASBIAS) val += brow[r];
                Ob[off] = val;
            }
        }
    }
}

// ---------------------------------------------------------------------------
// softmax over the 32-region axis of (B,32,N); optional hard-region argmax.
// ---------------------------------------------------------------------------
__global__ void softmax32_kernel(float* __restrict__ data, int* __restrict__ hr,
                                 int B, int N)
{
    long long idx = (long long)blockIdx.x * blockDim.x + threadIdx.x;
    long long total = (long long)B * N;
    if (idx >= total) return;
    int b = (int)(idx / N);
    int n = (int)(idx % N);
    float* col = data + (long long)b * 32 * N + n;
    float mx = -INFINITY; int am = 0;
    for (int k = 0; k < 32; ++k) {
        float v = col[(long long)k * N];
        if (v > mx) { mx = v; am = k; }
    }
    float s = 0.f;
    for (int k = 0; k < 32; ++k) {
        float e = __expf(col[(long long)k * N] - mx);
        col[(long long)k * N] = e;
        s += e;
    }
    float inv = 1.0f / s;
    for (int k = 0; k < 32; ++k) col[(long long)k * N] *= inv;
    if (hr) hr[(long long)b * N + n] = am;
}

// (B,Cdim,N) -> (B,N,Cdim) transpose, coalesced writes
__global__ void transpose_cn_kernel(const float* __restrict__ in, float* __restrict__ out,
                                    int B, int Cdim, int N)
{
    long long idx = (long long)blockIdx.x * blockDim.x + threadIdx.x;
    long long total = (long long)B * Cdim * N;
    if (idx >= total) return;
    int c = (int)(idx % Cdim);
    long long r = idx / Cdim;
    int n = (int)(r % N);
    int b = (int)(r / N);
    out[idx] = in[(long long)b * Cdim * N + (long long)c * N + n];
}

__global__ void zero_kernel(float* __restrict__ p, long long count)
{
    long long idx = (long long)blockIdx.x * blockDim.x + threadIdx.x;
    if (idx < count) p[idx] = 0.0f;
}

// ---------------------------------------------------------------------------
// Top-256 positions per (b,region): 256 threads, row in LDS, iterative argmax.
// Wave-shuffle argmax + single cross-wave LDS step (2 barriers/iteration).
// ---------------------------------------------------------------------------
__global__ __launch_bounds__(256)
void topk_pool_kernel(const float* __restrict__ sem, int* __restrict__ ti, int N)
{
    __shared__ float vals[4096];
    __shared__ float wV[8];
    __shared__ int   wI[8];
    const int tid  = threadIdx.x;
    const int lane = tid & 31;
    const int wv   = tid >> 5;
    const int br   = blockIdx.x;             // b*32 + r
    const float* row = sem + (long long)br * N;
    for (int i = tid; i < N; i += 256) vals[i] = row[i];
    __syncthreads();
    for (int t = 0; t < 256; ++t) {
        float best = -INFINITY; int bi = 0;
        for (int i = tid; i < N; i += 256) {
            float v = vals[i];
            if (v > best) { best = v; bi = i; }
        }
        for (int off = 16; off > 0; off >>= 1) {
            float ov = __shfl_xor(best, off, 32);
            int   oi = __shfl_xor(bi, off, 32);
            if (ov > best) { best = ov; bi = oi; }
        }
        if (lane == 0) { wV[wv] = best; wI[wv] = bi; }
        __syncthreads();
        if (tid == 0) {
            float bb = wV[0]; int ii = wI[0];
#pragma unroll
            for (int w = 1; w < 8; ++w)
                if (wV[w] > bb) { bb = wV[w]; ii = wI[w]; }
            ti[(long long)br * 256 + t] = ii;
            vals[ii] = -INFINITY;
        }
        __syncthreads();
    }
}

// ---------------------------------------------------------------------------
// Sparse top-k branch: one wave per position. q row cached in LDS, 8 pool
// entries per lane, wave argmax x8 for top-8, softmax, gather-V accumulate.
// Output stored position-major (B,N,C) for coalesced writes.
// ---------------------------------------------------------------------------
__global__ __launch_bounds__(256)
void sparse_attn_kernel(const float* __restrict__ qbuf,  // (B,64,N)
                        const float* __restrict__ kt,    // (B,N,64)
                        const float* __restrict__ vt,    // (B,N,256)
                        const int*  __restrict__ ti,     // (B,32,256)
                        const int*  __restrict__ hr,     // (B,N)
                        float* __restrict__ spT,         // (B,N,256)
                        int N, float invScale)
{
    __shared__ float qsh[8][64];
    const int wave = threadIdx.x >> 5;
    const int lane = threadIdx.x & 31;
    const long long gpos = (long long)blockIdx.x * 8 + wave;
    const int b = (int)(gpos / N);
    const int n = (int)(gpos % N);

    for (int j = lane; j < 64; j += 32)
        qsh[wave][j] = qbuf[((long long)b * 64 + j) * N + n];
    __syncthreads();

    const int r = hr[(long long)b * N + n];
    const int* pool = ti + ((long long)b * 32 + r) * 256;

    float sims[8];
#pragma unroll
    for (int j = 0; j < 8; ++j) {
        const int m = lane + 32 * j;
        const int p = pool[m];
        const float* krow = kt + ((long long)b * N + p) * 64;
        float s = 0.f;
#pragma unroll
        for (int dd = 0; dd < 64; ++dd) s += qsh[wave][dd] * krow[dd];
        sims[j] = s * invScale;
    }

    float topV[8]; int topP[8];
    unsigned used = 0u;
#pragma unroll
    for (int t = 0; t < 8; ++t) {
        float best = -INFINITY; int bm = -1;
#pragma unroll
        for (int j = 0; j < 8; ++j) {
            if (!((used >> j) & 1u) && sims[j] > best) { best = sims[j]; bm = lane + 32 * j; }
        }
        float bv = best; int bl = lane; int bmm = bm;
        for (int off = 16; off > 0; off >>= 1) {
            float ov = __shfl_xor(bv, off, 32);
            int   ol = __shfl_xor(bl, off, 32);
            int   om = __shfl_xor(bmm, off, 32);
            if (ov > bv || (ov == bv && ol < bl)) { bv = ov; bl = ol; bmm = om; }
        }
        if (lane == bl) used |= (1u << (bmm >> 5));
        topV[t] = bv;
        topP[t] = pool[bmm];
    }

    float att[8];
    float ssum = 0.f;
    const float mx = topV[0];
#pragma unroll
    for (int t = 0; t < 8; ++t) { att[t] = __expf(topV[t] - mx); ssum += att[t]; }
    const float inv = 1.0f / ssum;
#pragma unroll
    for (int t = 0; t < 8; ++t) att[t] *= inv;

#pragma unroll
    for (int j = 0; j < 8; ++j) {
        const int c = lane + 32 * j;
        float acc = 0.f;
#pragma unroll
        for (int t = 0; t < 8; ++t)
            acc += att[t] * vt[((long long)b * N + topP[t]) * 256 + c];
        spT[((long long)b * N + n) * 256 + c] = acc;
    }
}

// combined = sigmoid(alpha)*cont + (1-sig)*sparse   (in place over cont)
__global__ void combine_kernel(float* __restrict__ cont, const float* __restrict__ spT,
                               const float* __restrict__ alpha, int B, int C, int N)
{
    long long idx = (long long)blockIdx.x * blockDim.x + threadIdx.x;
    long long total = (long long)B * C * N;
    if (idx >= total) return;
    int n = (int)(idx % N);
    long long r = idx / N;
    int c = (int)(r % C);
    int b = (int)(r / C);
    float a = 1.0f / (1.0f + __expf(-alpha[0]));
    float cv = cont[idx];
    float sv = spT[((long long)b * N + n) * C + c];
    cont[idx] = a * cv + (1.0f - a) * sv;
}

// out = LayerNorm_C(x + fused) * gamma + beta ; one wave per position
__global__ __launch_bounds__(256)
void fuse_ln_kernel(const float* __restrict__ x, const float* __restrict__ fused,
                    const float* __restrict__ gamma, const float* __restrict__ beta,
                    float* __restrict__ out, int N)
{
    const int wave = threadIdx.x >> 5;
    const int lane = threadIdx.x & 31;
    const long long gpos = (long long)blockIdx.x * 8 + wave;
    const int b = (int)(gpos / N);
    const int n = (int)(gpos % N);

    float t[8];
    float s = 0.f, s2 = 0.f;
#pragma unroll
    for (int j = 0; j < 8; ++j) {
        const int c = lane + 32 * j;
        const long long off = ((long long)b * 256 + c) * N + n;
        float v = x[off] + fused[off];
        t[j] = v; s += v; s2 += v * v;
    }
    for (int off = 16; off > 0; off >>= 1) {
        s  += __shfl_xor(s, off, 32);
        s2 += __shfl_xor(s2, off, 32);
    }
    const float mu  = s * (1.0f / 256.0f);
    const float var = s2 * (1.0f / 256.0f) - mu * mu;
    const float inv = rsqrtf(var + 1e-5f);
#pragma unroll
    for (int j = 0; j < 8; ++j) {
        const int c = lane + 32 * j;
        out[((long long)b * 256 + c) * N + n] = (t[j] - mu) * inv * gamma[c] + beta[c];
    }
}

// ---------------------------------------------------------------------------
extern "C" void kernel_launch(void* const* d_in, const int* in_sizes, int n_in,
                              void* d_out, int out_size, void* d_ws, size_t ws_size,
                              hipStream_t stream)
{
    (void)in_sizes; (void)n_in; (void)out_size; (void)ws_size;
    const int B = 2, C = 256, N = 4096, D = 64, KSEM = 32, MPOOL = 256;
    const float scale = sqrtf((float)D) + 1e-6f;
    const float invScale = 1.0f / scale;

    const float* x      = (const float*)d_in[0];
    const float* sem_w  = (const float*)d_in[1];
    const float* sem_b  = (const float*)d_in[2];
    const float* q_w    = (const float*)d_in[3];
    const float* k_w    = (const float*)d_in[4];
    const float* v_w    = (const float*)d_in[5];
    const float* fuse_w = (const float*)d_in[6];
    const float* fuse_b = (const float*)d_in[7];
    const float* alpha  = (const float*)d_in[8];
    const float* ln_g   = (const float*)d_in[9];
    const float* ln_b   = (const float*)d_in[10];
    float* outp = (float*)d_out;

    float* w = (float*)d_ws;
    size_t off = 0;
    float* sem  = w + off; off += (size_t)B * KSEM * N;   // sem logits -> softmax in place
    float* qb   = w + off; off += (size_t)B * D * N;
    float* kb   = w + off; off += (size_t)B * D * N;
    float* vb   = w + off; off += (size_t)B * C * N;
    float* kt   = w + off; off += (size_t)B * N * D;
    float* vt   = w + off; off += (size_t)B * N * C;
    float* regk = w + off; off += (size_t)B * KSEM * D;
    float* regv = w + off; off += (size_t)B * KSEM * C;
    float* attn = w + off; off += (size_t)B * KSEM * N;
    float* cont = w + off; off += (size_t)B * C * N;      // also holds `combined`
    float* spT  = w + off; off += (size_t)B * N * C;
    float* fus  = w + off; off += (size_t)B * C * N;
    int* ti = (int*)(w + off); off += (size_t)B * KSEM * MPOOL;
    int* hr = (int*)(w + off); off += (size_t)B * N;

    // 1-4: projections  W (Mx256) @ xf (256x4096)
    wmma_gemm_f32<false, true, true><<<dim3(B, KSEM / 16, N / 64), 32, 0, stream>>>(
        sem_w, 0, C, 1, x, (long long)C * N, sem_b, sem, (long long)KSEM * N, C, N, N / 64, 1, 1.0f);
    wmma_gemm_f32<false, false, true><<<dim3(B, D / 16, N / 64), 32, 0, stream>>>(
        q_w, 0, C, 1, x, (long long)C * N, nullptr, qb, (long long)D * N, C, N, N / 64, 1, 1.0f);
    wmma_gemm_f32<false, false, true><<<dim3(B, D / 16, N / 64), 32, 0, stream>>>(
        k_w, 0, C, 1, x, (long long)C * N, nullptr, kb, (long long)D * N, C, N, N / 64, 1, 1.0f);
    wmma_gemm_f32<false, false, true><<<dim3(B, C / 16, N / 64), 32, 0, stream>>>(
        v_w, 0, C, 1, x, (long long)C * N, nullptr, vb, (long long)C * N, C, N, N / 64, 1, 1.0f);

    // 5: region softmax + hard assignment
    softmax32_kernel<<<(B * N + 255) / 256, 256, 0, stream>>>(sem, hr, B, N);

    // 6-7: transposes for contiguous gathers / region GEMMs
    transpose_cn_kernel<<<(int)(((long long)B * D * N + 255) / 256), 256, 0, stream>>>(kb, kt, B, D, N);
    transpose_cn_kernel<<<(int)(((long long)B * C * N + 255) / 256), 256, 0, stream>>>(vb, vt, B, C, N);

    // 8: zero split-K accumulators (regk and regv are adjacent)
    zero_kernel<<<(int)(((long long)B * KSEM * (D + C) + 255) / 256), 256, 0, stream>>>(
        regk, (long long)B * KSEM * (D + C));

    // 9: region_k = sem (32x4096) @ kt (4096x64), split-K=16
    wmma_gemm_f32<true, false, true><<<dim3(B, KSEM / 16, 1 * 16), 32, 0, stream>>>(
        sem, (long long)KSEM * N, N, 1, kt, (long long)N * D, nullptr,
        regk, (long long)KSEM * D, N, D, 1, 16, 1.0f);
    // 10: region_v = sem (32x4096) @ vt (4096x256), split-K=16
    wmma_gemm_f32<true, false, true><<<dim3(B, KSEM / 16, (C / 64) * 16), 32, 0, stream>>>(
        sem, (long long)KSEM * N, N, 1, vt, (long long)N * C, nullptr,
        regv, (long long)KSEM * C, N, C, C / 64, 16, 1.0f);

    // 11: scores = region_k (32x64) @ q (64x4096) / scale
    wmma_gemm_f32<false, false, true><<<dim3(B, KSEM / 16, N / 64), 32, 0, stream>>>(
        regk, (long long)KSEM * D, D, 1, qb, (long long)D * N, nullptr,
        attn, (long long)KSEM * N, D, N, N / 64, 1, invScale);
    // 12: softmax over regions
    softmax32_kernel<<<(B * N + 255) / 256, 256, 0, stream>>>(attn, nullptr, B, N);

    // 13: cont = region_v^T (256x32) @ attn (32x4096)   (A strided = transposed)
    wmma_gemm_f32<false, false, false><<<dim3(B, C / 16, N / 64), 32, 0, stream>>>(
        regv, (long long)KSEM * C, 1, C, attn, (long long)KSEM * N, nullptr,
        cont, (long long)C * N, KSEM, N, N / 64, 1, 1.0f);

    // 14: top-256 pool indices per (b, region)
    topk_pool_kernel<<<B * KSEM, 256, 0, stream>>>(sem, ti, N);

    // 15: sparse gather attention (wave per position)
    sparse_attn_kernel<<<B * N / 8, 256, 0, stream>>>(qb, kt, vt, ti, hr, spT, N, invScale);

    // 16: combined = a*cont + (1-a)*sparse  (in place into cont)
    combine_kernel<<<(int)(((long long)B * C * N + 255) / 256), 256, 0, stream>>>(
        cont, spT, alpha, B, C, N);

    // 17: fused = fuse_w (256x256) @ combined (256x4096) + fuse_b
    wmma_gemm_f32<false, true, true><<<dim3(B, C / 16, N / 64), 32, 0, stream>>>(
        fuse_w, 0, C, 1, cont, (long long)C * N, fuse_b, fus, (long long)C * N, C, N, N / 64, 1, 1.0f);

    // 18: residual + channel LayerNorm -> output
    fuse_ln_kernel<<<B * N / 8, 256, 0, stream>>>(x, fus, ln_g, ln_b, outp, N);
}